// Projector_27960237097267
// MI455X (gfx1250) — compile-verified
//
#include <hip/hip_runtime.h>

#define ZDIM 64
#define YDIM 512
#define XDIM 512
#define NANG 720
#define NCOL 512
#define TSTEPS 725
#define THALF 362.0f     // (T-1)/2
#define CXY 255.5f       // (X-1)/2 == (Y-1)/2

// ---------------------------------------------------------------------------
// Pre-pass: transpose vol [Z,Y,X] -> [Y,X,Z] so that z (the lane dimension of
// the main kernel) is contiguous. Fully coalesced loads and stores via LDS.
// ---------------------------------------------------------------------------
__global__ __launch_bounds__(256)
void transpose_zyx_to_yxz(const float* __restrict__ vol, float* __restrict__ out) {
    __shared__ float tile[ZDIM][32 + 1];
    const int y  = blockIdx.x;        // 0..511
    const int x0 = blockIdx.y * 32;   // 16 tiles of 32 columns

    for (int i = threadIdx.x; i < ZDIM * 32; i += 256) {
        const int z = i >> 5;
        const int x = i & 31;
        tile[z][x] = vol[(size_t)z * (YDIM * XDIM) + (size_t)y * XDIM + (x0 + x)];
    }
    __syncthreads();
    for (int i = threadIdx.x; i < ZDIM * 32; i += 256) {
        const int z = i & 63;
        const int x = i >> 6;
        out[((size_t)y * XDIM + (x0 + x)) * ZDIM + z] = tile[z][x];
    }
}

// Load a z-pair (zb, zb+1) for one bilinear corner. In the transposed layout
// (SZ==1) this is a single 8-byte load -> global_load_b64 from scalar base.
template<int SZ>
__device__ __forceinline__ float2 ld2(const float* __restrict__ p, int voff) {
    if constexpr (SZ == 1) {
        return *reinterpret_cast<const float2*>(p + voff);
    } else {
        float2 r;
        r.x = p[voff];
        r.y = p[voff + SZ];
        return r;
    }
}

// ---------------------------------------------------------------------------
// Main kernel. One wave32 per (angle, s); each lane owns z = 2*lane, 2*lane+1,
// so one wave produces all 64 z outputs of a sinogram column. All ray math is
// lane-uniform -> promoted to SGPRs via readfirstlane; the 4 bilinear corner
// gathers are global_load_b64 saddr + lane*8 + imm (two aligned 128B lines).
// Template strides (elements): SY = y stride, SX = x stride, SZ = z stride.
// ---------------------------------------------------------------------------
template<int SY, int SX, int SZ>
__global__ __launch_bounds__(256)
void radon_fwd(const float* __restrict__ vol, const float* __restrict__ phis,
               float* __restrict__ out) {
    const int lane = threadIdx.x & 31;
    const int wave = threadIdx.x >> 5;
    const int ang  = blockIdx.x;
    const int s    = blockIdx.y * 8 + wave;   // detector column 0..511
    const int zb   = 2 * lane;                // first z of this lane's pair
    const int voff = zb * SZ;

    const float phi = phis[ang];
    const float c   = cosf(phi);
    const float sn  = sinf(phi);
    const float sv  = (float)s - CXY;

    // x(k) = X0 - k*sn ; y(k) = Y0 + k*c   with k = 0..T-1 (t = k - THALF)
    const float X0 = fmaf(sv, c,  fmaf( THALF, sn, CXY));
    const float Y0 = fmaf(sv, sn, fmaf(-THALF, c,  CXY));

    // Conservative chord clipping: contribution possible only while
    // x in (-1, X) and y in (-1, Y). Expanded by 1 step; per-step masks below
    // make this safe.
    float lo = 0.f, hi = (float)(TSTEPS - 1);
    if (sn > 1e-6f)       { lo = fmaxf(lo, (X0 - (float)XDIM) / sn); hi = fminf(hi, (X0 + 1.f) / sn); }
    else if (sn < -1e-6f) { lo = fmaxf(lo, (X0 + 1.f) / sn);         hi = fminf(hi, (X0 - (float)XDIM) / sn); }
    else if (X0 <= -1.f || X0 >= (float)XDIM) { hi = -1.f; }
    if (c > 1e-6f)        { lo = fmaxf(lo, (-1.f - Y0) / c);         hi = fminf(hi, ((float)YDIM - Y0) / c); }
    else if (c < -1e-6f)  { lo = fmaxf(lo, ((float)YDIM - Y0) / c);  hi = fminf(hi, (-1.f - Y0) / c); }
    else if (Y0 <= -1.f || Y0 >= (float)YDIM) { hi = -1.f; }

    const int klo = max(0, (int)floorf(lo) - 1);
    const int khi = min(TSTEPS - 1, (int)ceilf(hi) + 1);

    // address delta per ray step (elements), for prefetching ~8 steps ahead
    const int pfd = __builtin_amdgcn_readfirstlane(
        (int)(8.f * (c * (float)SY - sn * (float)SX)));

    float2 acc;
    acc.x = 0.f;
    acc.y = 0.f;

#pragma unroll 2
    for (int k = klo; k <= khi; ++k) {
        const float kf = (float)k;
        const float x  = fmaf(kf, -sn, X0);
        const float y  = fmaf(kf,  c,  Y0);
        const float fx = floorf(x);
        const float fy = floorf(y);
        const float wx = x - fx;
        const float wy = y - fy;
        const int xs = __builtin_amdgcn_readfirstlane((int)fx);
        const int ys = __builtin_amdgcn_readfirstlane((int)fy);

        if (xs >= 0 && xs < XDIM - 1 && ys >= 0 && ys < YDIM - 1) {
            // fast interior path: 4 coalesced b64 gathers from one scalar base
            const float* q = vol + ((size_t)ys * SY + (size_t)xs * SX);
            __builtin_prefetch(q + voff + pfd, 0, 3);  // global_prefetch_b8
            const float2 g00 = ld2<SZ>(q, voff);
            const float2 g01 = ld2<SZ>(q + SX, voff);
            const float2 g10 = ld2<SZ>(q + SY, voff);
            const float2 g11 = ld2<SZ>(q + SY + SX, voff);
            const float gx0a = fmaf(wx, g01.x - g00.x, g00.x);
            const float gx1a = fmaf(wx, g11.x - g10.x, g10.x);
            const float gx0b = fmaf(wx, g01.y - g00.y, g00.y);
            const float gx1b = fmaf(wx, g11.y - g10.y, g10.y);
            acc.x += fmaf(wy, gx1a - gx0a, gx0a);
            acc.y += fmaf(wy, gx1b - gx0b, gx0b);
        } else {
            // boundary path (rare, uniform): clamp + per-corner validity mask
            const bool vx0 = (xs >= 0)     && (xs < XDIM);
            const bool vx1 = (xs + 1 >= 0) && (xs + 1 < XDIM);
            const bool vy0 = (ys >= 0)     && (ys < YDIM);
            const bool vy1 = (ys + 1 >= 0) && (ys + 1 < YDIM);
            const int xc0 = min(max(xs, 0), XDIM - 1);
            const int xc1 = min(max(xs + 1, 0), XDIM - 1);
            const int yc0 = min(max(ys, 0), YDIM - 1);
            const int yc1 = min(max(ys + 1, 0), YDIM - 1);
            float2 zero; zero.x = 0.f; zero.y = 0.f;
            const float2 g00 = (vx0 && vy0) ? ld2<SZ>(vol + (size_t)yc0 * SY + (size_t)xc0 * SX, voff) : zero;
            const float2 g01 = (vx1 && vy0) ? ld2<SZ>(vol + (size_t)yc0 * SY + (size_t)xc1 * SX, voff) : zero;
            const float2 g10 = (vx0 && vy1) ? ld2<SZ>(vol + (size_t)yc1 * SY + (size_t)xc0 * SX, voff) : zero;
            const float2 g11 = (vx1 && vy1) ? ld2<SZ>(vol + (size_t)yc1 * SY + (size_t)xc1 * SX, voff) : zero;
            const float w00 = (1.f - wy) * (1.f - wx);
            const float w01 = (1.f - wy) * wx;
            const float w10 = wy * (1.f - wx);
            const float w11 = wy * wx;
            acc.x += w00 * g00.x + w01 * g01.x + w10 * g10.x + w11 * g11.x;
            acc.y += w00 * g00.y + w01 * g01.y + w10 * g10.y + w11 * g11.y;
        }
    }

    // ---- epilogue: out [1, 720, 64, 512]; element (ang, zb..zb+1, s) -------
    float* g0 = out + (size_t)ang * (ZDIM * NCOL) + (size_t)zb * NCOL + s;

#if defined(__HIP_DEVICE_COMPILE__) && __has_builtin(__builtin_amdgcn_global_store_async_from_lds_b32)
    // Stage results through LDS and drain with async LDS->global stores
    // (ASYNCcnt path). Once per wave: zero inner-loop cost.
    typedef __attribute__((address_space(1))) int gint_t;
    typedef __attribute__((address_space(3))) int lint_t;
    __shared__ float sacc[8][ZDIM];
    sacc[wave][zb]     = acc.x;
    sacc[wave][zb + 1] = acc.y;
    asm volatile("s_wait_dscnt 0x0" ::: "memory");  // LDS write visible to async unit
    __builtin_amdgcn_global_store_async_from_lds_b32(
        (gint_t*)g0, (lint_t*)&sacc[wave][zb], 0, 0);
    __builtin_amdgcn_global_store_async_from_lds_b32(
        (gint_t*)(g0 + NCOL), (lint_t*)&sacc[wave][zb + 1], 0, 0);
    asm volatile("s_wait_asynccnt 0x0" ::: "memory");
#else
    g0[0]    = acc.x;
    g0[NCOL] = acc.y;
#endif
}

// ---------------------------------------------------------------------------
extern "C" void kernel_launch(void* const* d_in, const int* in_sizes, int n_in,
                              void* d_out, int out_size, void* d_ws, size_t ws_size,
                              hipStream_t stream) {
    const float* vol  = (const float*)d_in[0];
    const float* phis = (const float*)d_in[1];
    float* out = (float*)d_out;

    const size_t need = (size_t)YDIM * XDIM * ZDIM * sizeof(float);
    const dim3 grid(NANG, NCOL / 8);   // 720 x 64 blocks, 8 waves/block

    if (ws_size >= need) {
        // transpose into z-contiguous layout, then lane-coalesced b64 gathers
        float* vt = (float*)d_ws;
        transpose_zyx_to_yxz<<<dim3(YDIM, XDIM / 32), 256, 0, stream>>>(vol, vt);
        radon_fwd<XDIM * ZDIM, ZDIM, 1><<<grid, 256, 0, stream>>>(vt, phis, out);
    } else {
        // fallback: original [Z,Y,X] layout (z stride = Y*X)
        radon_fwd<XDIM, 1, YDIM * XDIM><<<grid, 256, 0, stream>>>(vol, phis, out);
    }
}